// VectorKuramoto_8229157339889
// MI455X (gfx1250) — compile-verified
//
#include <hip/hip_runtime.h>
#include <math.h>

// Problem constants (reference: B=4, N=2048, D=4)
#define BN 4
#define NN 2048
#define DD 4

__device__ __constant__ float kCOUPLING  = 1.0f;
__device__ __constant__ float kDT        = 1.0f;
__device__ __constant__ float kATTRACT   = 1.0f;
__device__ __constant__ float kEPS       = 1e-6f;

typedef float v2f __attribute__((ext_vector_type(2)));
typedef float v4f __attribute__((ext_vector_type(4)));
typedef float v8f __attribute__((ext_vector_type(8)));

// ---------------------------------------------------------------------------
// Pre-pass: sc[b][j][0..3] = sin(theta[b][j][d]), sc[b][j][4..7] = cos(...)
// 256 KB table in d_ws; computed once with precise libm sin/cos.
// ---------------------------------------------------------------------------
__global__ void vk_sc_kernel(const float* __restrict__ theta,
                             float* __restrict__ sc) {
    int t = blockIdx.x * blockDim.x + threadIdx.x;   // one thread per (b, j)
    if (t >= BN * NN) return;
    const float4 th = ((const float4*)theta)[t];
    float4 s, c;
    s.x = sinf(th.x); c.x = cosf(th.x);
    s.y = sinf(th.y); c.y = cosf(th.y);
    s.z = sinf(th.z); c.z = cosf(th.z);
    s.w = sinf(th.w); c.w = cosf(th.w);
    ((float4*)sc)[2 * t + 0] = s;
    ((float4*)sc)[2 * t + 1] = c;
}

// ---------------------------------------------------------------------------
// Main kernel: 64 threads (2 waves), each wave owns one 16-row WMMA tile.
// grid.x = BN * (NN/32) = 256 blocks.
//
// Streams W and alpha exactly once (134 MB total -> HBM-bound) with
// non-temporal b128 loads, reduces with v_wmma_f32_16x16x4_f32 (f32-exact).
//
// K-slot reordering: per j-iteration (step 8), WMMA#1 carries K-slots
// {j, j+1, j+4, j+5} and WMMA#2 carries {j+2, j+3, j+6, j+7}; lane-half h
// supplies slots from its float4 at column j+4h, and the B operand (from the
// LDS sin/cos table) uses the identical slot->row mapping, so the full j-sum
// is exact regardless of slot order.
//
// B columns 8-15 intentionally duplicate columns 0-7 (finite values); the
// epilogue only reads accumulator columns 0-7, so no masking is needed and
// EXEC stays all-ones for every WMMA.
// ---------------------------------------------------------------------------
__launch_bounds__(64)
__global__ void vk_wmma_kernel(const float* __restrict__ theta,
                               const float* __restrict__ gamma,
                               const float* __restrict__ W,
                               const float* __restrict__ alpha,
                               const float* __restrict__ sc,
                               float* __restrict__ out) {
    __shared__ float lds_sc[NN * 8];   // 64 KB: [s0..3 c0..3] per row of batch

    const int b       = blockIdx.x >> 6;      // NN/32 = 64 blocks per batch
    const int tileIdx = blockIdx.x & 63;      // 32-row block within batch
    const int tid     = threadIdx.x;
    const int wave    = tid >> 5;             // 0..1
    const int lane    = tid & 31;

    // ---- stage this batch's sin/cos table into LDS (cooperative, 64 KB) ----
    {
        const float4* src = (const float4*)(sc + (size_t)b * NN * 8);
        float4*       dst = (float4*)lds_sc;
        for (int i = tid; i < NN * 2; i += 64) dst[i] = src[i];
    }
    __syncthreads();

    // ---- A-matrix (16x4 f32) lane mapping: lanes 0-15 slots 0,1; 16-31 2,3 -
    const int row   = tileIdx * 32 + wave * 16 + (lane & 15); // row in batch
    const int khalf = lane >> 4;                              // 0 or 1
    const int col   = lane & 15;                              // B/D column
    const int c7    = col & 7;

    const size_t rowOff = ((size_t)b * NN + row) * NN;
    const v4f* wrow = (const v4f*)(W + rowOff);
    const v4f* arow = (const v4f*)(alpha + rowOff);

    v8f acc0 = {};   // [ A ·s (cols0-3) | A ·c (cols4-7) | dup ]
    v8f acc1 = {};   // [ Bm·s (cols0-3) | Bm·c (cols4-7) | dup ]

#pragma unroll 4
    for (int j = 0; j < NN; j += 8) {
        const int f4idx = (j >> 2) + khalf;           // float4 index: j + 4h
        const v4f w4 = __builtin_nontemporal_load(&wrow[f4idx]);
        const v4f a4 = __builtin_nontemporal_load(&arow[f4idx]);

        const float ca0 = __cosf(a4.x), sa0 = __sinf(a4.x);
        const float ca1 = __cosf(a4.y), sa1 = __sinf(a4.y);
        const float ca2 = __cosf(a4.z), sa2 = __sinf(a4.z);
        const float ca3 = __cosf(a4.w), sa3 = __sinf(a4.w);

        v2f A1 = { w4.x * ca0, w4.y * ca1 };          // W*cos: slots of WMMA#1
        v2f A2 = { w4.z * ca2, w4.w * ca3 };          //        slots of WMMA#2
        v2f M1 = { w4.x * sa0, w4.y * sa1 };          // W*sin
        v2f M2 = { w4.z * sa2, w4.w * sa3 };

        // B operand rows: sc[j+4h+t][col], t=0..3 (cols 8-15 = benign dups)
        const int base = (j + 4 * khalf) * 8 + c7;
        v2f B1 = { lds_sc[base],      lds_sc[base + 8]  };
        v2f B2 = { lds_sc[base + 16], lds_sc[base + 24] };

        acc0 = __builtin_amdgcn_wmma_f32_16x16x4_f32(
            false, A1, false, B1, (short)0, acc0, false, false);
        acc1 = __builtin_amdgcn_wmma_f32_16x16x4_f32(
            false, M1, false, B1, (short)0, acc1, false, false);
        acc0 = __builtin_amdgcn_wmma_f32_16x16x4_f32(
            false, A2, false, B2, (short)0, acc0, false, false);
        acc1 = __builtin_amdgcn_wmma_f32_16x16x4_f32(
            false, M2, false, B2, (short)0, acc1, false, false);
    }

    // ---- epilogue: C/D layout = lane(0-15:N, M=v; 16-31:N, M=v+8) ----------
    const int  d       = lane & 3;
    const bool active  = (lane & 12) == 0;     // lanes 0-3 and 16-19
    const int  srcbase = lane & 16;
    const float scale  = kCOUPLING / (float)NN;

#pragma unroll
    for (int m = 0; m < 8; ++m) {
        const float As = __shfl(acc0[m], srcbase + d,     32);
        const float Ac = __shfl(acc0[m], srcbase + d + 4, 32);
        const float Bs = __shfl(acc1[m], srcbase + d,     32);
        const float Bc = __shfl(acc1[m], srcbase + d + 4, 32);

        const int i = tileIdx * 32 + wave * 16 + m + ((lane >> 4) << 3);
        const float si = lds_sc[i * 8 + d];
        const float ci = lds_sc[i * 8 + 4 + d];

        const size_t gidx = ((size_t)b * NN + i) * DD + d;
        const float th = theta[gidx];
        const float gm = gamma[gidx];

        const float coup = scale * (ci * As - si * Ac - ci * Bc - si * Bs);
        float tn = th + kDT * (kATTRACT * (gm - th) + coup);

        // L2 norm over the 4 components (quad reduction; quads 0-3 / 16-19)
        float n2 = tn * tn;
        n2 += __shfl_xor(n2, 1, 32);
        n2 += __shfl_xor(n2, 2, 32);
        const float nrm = fmaxf(sqrtf(n2), kEPS);

        if (active) out[gidx] = tn / nrm;
    }
}

// ---------------------------------------------------------------------------
extern "C" void kernel_launch(void* const* d_in, const int* in_sizes, int n_in,
                              void* d_out, int out_size, void* d_ws, size_t ws_size,
                              hipStream_t stream) {
    const float* theta = (const float*)d_in[0];   // [B,N,D]
    const float* gamma = (const float*)d_in[1];   // [B,N,D]
    const float* W     = (const float*)d_in[2];   // [B,N,N]
    const float* alpha = (const float*)d_in[3];   // [B,N,N]
    float* out = (float*)d_out;                   // [B,N,D]
    float* sc  = (float*)d_ws;                    // B*N*8 floats = 256 KB

    vk_sc_kernel<<<(BN * NN + 255) / 256, 256, 0, stream>>>(theta, sc);
    vk_wmma_kernel<<<BN * (NN / 32), 64, 0, stream>>>(theta, gamma, W, alpha,
                                                      sc, out);
}